// Router_70446053589280
// MI455X (gfx1250) — compile-verified
//
#include <hip/hip_runtime.h>

// MoE router for MI455X (gfx1250, wave32, WMMA + TDM).
// logits = X[16384,4096] @ W[64,4096]^T ; softmax over 64 experts; top-2.
// Bandwidth-bound: 256MB activation stream @ 23.3TB/s ~= 11us floor, right at
// the f32 WMMA (16x16x4) roofline. Staging uses the Tensor Data Mover:
// one TENSOR_LOAD_TO_LDS per tile (16 rows x 512B, row stride 16KB), with the
// D# pad feature (128 DWORDs -> +4 DWORDs) reproducing the 132-float LDS
// stride that makes the WMMA fragment b64 reads bank-conflict-free.
// Double-buffered: TDM DMA for chunk c+1 overlaps the WMMA sweep of chunk c.
// Fragment loads are software-pipelined one k-group ahead to hide LDS latency.

typedef __attribute__((ext_vector_type(2))) float        v2f;
typedef __attribute__((ext_vector_type(8))) float        v8f;
typedef __attribute__((ext_vector_type(2))) int          v2i;
typedef __attribute__((ext_vector_type(4))) int          v4i;
typedef __attribute__((ext_vector_type(8))) int          v8i;
typedef __attribute__((ext_vector_type(4))) unsigned int v4u;

#define AS1 __attribute__((address_space(1)))
#define AS3 __attribute__((address_space(3)))

#if __has_builtin(__builtin_amdgcn_tensor_load_to_lds)
#define USE_TDM 1
#else
#define USE_TDM 0
#endif
#if __has_builtin(__builtin_amdgcn_global_load_async_to_lds_b64)
#define USE_ASYNC_LDS 1
#else
#define USE_ASYNC_LDS 0
#endif

#define HIDDEN     4096
#define NEXP       64
#define KC         128                  // K chunk per staging round
#define NCHUNK     (HIDDEN / KC)        // 32
#define BSTRIDE    132                  // stride % 64 == 4 -> conflict-free b64 frag reads
#define ASTRIDE    132
#define B_FLOATS   (NEXP * BSTRIDE)     // 8448 floats
#define AWAVE      (16 * ASTRIDE)       // 2112 floats per wave tile
#define A_FLOATS   (4 * AWAVE)          // 8448 floats
#define BUF_FLOATS (B_FLOATS + A_FLOATS)    // 16896 floats per buffer
#define BUF_BYTES  (BUF_FLOATS * 4)
#define SMEM_FLOATS (2 * BUF_FLOATS)        // 135168 B total (2 bufs)
#define LOGSTRIDE  65

__device__ __forceinline__ void wait_async0() {
#if __has_builtin(__builtin_amdgcn_s_wait_asynccnt)
    __builtin_amdgcn_s_wait_asynccnt(0);
#else
    asm volatile("s_wait_asynccnt 0" ::: "memory");
#endif
}
__device__ __forceinline__ void wait_tensor0() {
#if __has_builtin(__builtin_amdgcn_s_wait_tensorcnt)
    __builtin_amdgcn_s_wait_tensorcnt(0);
#else
    asm volatile("s_wait_tensorcnt 0" ::: "memory");
#endif
}

#if USE_TDM
// D# group0: [1:0]=count=1, [63:32]=lds byte addr, [120:64]=global addr,
// [127:126]=type=2.
__device__ __forceinline__ v4u tdm_g0(unsigned long long gaddr, unsigned lds) {
    v4u g0;
    g0.x = 1u;
    g0.y = lds;
    g0.z = (unsigned)gaddr;
    g0.w = (unsigned)(gaddr >> 32) | 0x80000000u;   // type=2
    return g0;
}
__device__ __forceinline__ void tdm_load(v4u g0, v8i g1) {
    v4i z4 = {0, 0, 0, 0};
#if __clang_major__ >= 23
    v8i z8 = {0, 0, 0, 0, 0, 0, 0, 0};
    __builtin_amdgcn_tensor_load_to_lds(g0, g1, z4, z4, z8, 0);
#else
    __builtin_amdgcn_tensor_load_to_lds(g0, g1, z4, z4, 0);
#endif
}
#endif

// Non-TDM staging of one K-chunk (async-to-LDS or plain VGPR fallback).
__device__ __forceinline__ void stage_chunk(const float* __restrict__ X,
                                            const float* __restrict__ W,
                                            float* buf, int k0,
                                            int tid, int wave, int lane,
                                            int tok0_w)
{
    float* bufB = buf;
    float* bufA = buf + B_FLOATS + wave * AWAVE;
#if USE_ASYNC_LDS
    for (int i = 0; i < 32; ++i) {                 // B: 64 rows x 64 b64
        int idx = i * 128 + tid;
        int row = idx >> 6;
        int c2  = (idx & 63) * 2;
        __builtin_amdgcn_global_load_async_to_lds_b64(
            (AS1 v2i*)(W + (size_t)row * HIDDEN + k0 + c2),
            (AS3 v2i*)(bufB + row * BSTRIDE + c2), 0, 0);
    }
    for (int i = 0; i < 32; ++i) {                 // A: 16 rows x 64 b64
        int idx = i * 32 + lane;
        int row = idx >> 6;
        int c2  = (idx & 63) * 2;
        __builtin_amdgcn_global_load_async_to_lds_b64(
            (AS1 v2i*)(X + (size_t)(tok0_w + row) * HIDDEN + k0 + c2),
            (AS3 v2i*)(bufA + row * ASTRIDE + c2), 0, 0);
    }
#else
    for (int i = 0; i < 16; ++i) {
        int idx = i * 128 + tid;
        int row = idx >> 5;
        int c4  = idx & 31;
        float4 v = reinterpret_cast<const float4*>(W + (size_t)row * HIDDEN + k0)[c4];
        float* dst = bufB + row * BSTRIDE + c4 * 4;
        v2f lo = {v.x, v.y};
        v2f hi = {v.z, v.w};
        *reinterpret_cast<v2f*>(dst)     = lo;
        *reinterpret_cast<v2f*>(dst + 2) = hi;
    }
    for (int r = 0; r < 16; ++r) {
        float4 v = reinterpret_cast<const float4*>(
            X + (size_t)(tok0_w + r) * HIDDEN + k0)[lane];
        float* dst = bufA + r * ASTRIDE + lane * 4;
        v2f lo = {v.x, v.y};
        v2f hi = {v.z, v.w};
        *reinterpret_cast<v2f*>(dst)     = lo;
        *reinterpret_cast<v2f*>(dst + 2) = hi;
    }
#endif
}

__global__ __launch_bounds__(128) void router_topk_kernel(
    const float* __restrict__ X, const float* __restrict__ W,
    float* __restrict__ out_probs, float* __restrict__ out_idx)
{
    extern __shared__ float smem[];
    const int tid  = threadIdx.x;
    const int wave = tid >> 5;
    const int lane = tid & 31;
    const int half = lane >> 4;
    const int l16  = lane & 15;

    const int tok0_wg = blockIdx.x * 64;
    const int tok0_w  = tok0_wg + wave * 16;

#if USE_TDM
    // Wave-uniform (SGPR) descriptor state. Each wave DMAs its own A tile
    // (16 tokens x 128) and a 16-expert quarter of B -- identical shapes, so
    // one shared group1 descriptor:
    //   data_size=4B, pad: every 128 DWORDs insert 4 (-> stride 132 floats),
    //   tensor_dim0=4096, tensor_dim1=16, tile=128x16, dim0_stride=4096.
    const int uwave = __builtin_amdgcn_readfirstlane(wave);
    const unsigned long long gaddrB =
        (unsigned long long)(const void*)W + (unsigned long long)uwave * 16ull * HIDDEN * 4ull;
    const unsigned long long gaddrA =
        (unsigned long long)(const void*)X +
        ((unsigned long long)tok0_wg + (unsigned long long)uwave * 16ull) * HIDDEN * 4ull;
    const unsigned ldsB0 = (unsigned)(size_t)(AS3 float*)(smem + uwave * 16 * BSTRIDE);
    const unsigned ldsA0 = (unsigned)(size_t)(AS3 float*)(smem + B_FLOATS + uwave * AWAVE);
    const v8i g1 = { (int)((2u << 16) | (1u << 20) | (6u << 22) | (3u << 25)),
                     (int)(4096u << 16),   // tensor_dim0[15:0] @ [31:16]
                     (int)(16u << 16),     // dim0 hi16=0 | tensor_dim1 lo16
                     (int)(128u << 16),    // dim1 hi16=0 | tile_dim0=128
                     16,                   // tile_dim1=16, tile_dim2=0
                     4096,                 // tensor_dim0_stride lo32
                     0, 0 };
    // Prologue: DMA chunk 0 into buffer 0.
    tdm_load(tdm_g0(gaddrB, ldsB0), g1);
    tdm_load(tdm_g0(gaddrA, ldsA0), g1);
    wait_tensor0();
#else
    stage_chunk(X, W, smem, 0, tid, wave, lane, tok0_w);
#if USE_ASYNC_LDS
    wait_async0();
#endif
#endif
    __syncthreads();

    v8f acc0 = {}, acc1 = {}, acc2 = {}, acc3 = {};

    for (int c = 0; c < NCHUNK; ++c) {
        float* cur = smem + (c & 1) * BUF_FLOATS;

        // Kick off DMA for the next chunk into the other buffer; TENSORcnt /
        // ASYNCcnt traffic is unordered vs this wave's ds_loads, so the DMA
        // overlaps the WMMA sweep below.
        if (c + 1 < NCHUNK) {
#if USE_TDM
            unsigned long long koff = (unsigned long long)(c + 1) * (KC * 4);
            unsigned sel = (unsigned)((c + 1) & 1) * BUF_BYTES;
            tdm_load(tdm_g0(gaddrB + koff, ldsB0 + sel), g1);
            tdm_load(tdm_g0(gaddrA + koff, ldsA0 + sel), g1);
#else
            stage_chunk(X, W, smem + ((c + 1) & 1) * BUF_FLOATS, (c + 1) * KC,
                        tid, wave, lane, tok0_w);
#endif
        }

        // ---- WMMA sweep: D[16 tok x 16 exp] += A(16x4)*B(4x16), 4 expert tiles.
        // A frag: lanes 0-15 hold K={k,k+1}, lanes 16-31 K={k+2,k+3} of row l16.
        // B frag: lane n holds W[e0+n][k..k+1] (+2 upper half) = B[k][n].
        // Software-pipelined one k-group ahead so the s_wait_dscnt for group g
        // lands behind group g-1's WMMA issue window.
        const float* aRow = cur + B_FLOATS + wave * AWAVE + l16 * ASTRIDE + half * 2;
        const float* bRow = cur + l16 * BSTRIDE + half * 2;
        v2f a  = *reinterpret_cast<const v2f*>(aRow);
        v2f b0 = *reinterpret_cast<const v2f*>(bRow);
        v2f b1 = *reinterpret_cast<const v2f*>(bRow + 16 * BSTRIDE);
        v2f b2 = *reinterpret_cast<const v2f*>(bRow + 32 * BSTRIDE);
        v2f b3 = *reinterpret_cast<const v2f*>(bRow + 48 * BSTRIDE);
        #pragma unroll 8
        for (int kk = 0; kk < KC; kk += 4) {
            const int nk = (kk + 4 < KC) ? kk + 4 : kk;   // tail reloads self
            v2f na  = *reinterpret_cast<const v2f*>(aRow + nk);
            v2f nb0 = *reinterpret_cast<const v2f*>(bRow + nk);
            v2f nb1 = *reinterpret_cast<const v2f*>(bRow + 16 * BSTRIDE + nk);
            v2f nb2 = *reinterpret_cast<const v2f*>(bRow + 32 * BSTRIDE + nk);
            v2f nb3 = *reinterpret_cast<const v2f*>(bRow + 48 * BSTRIDE + nk);
            acc0 = __builtin_amdgcn_wmma_f32_16x16x4_f32(false, a, false, b0, (short)0, acc0, false, false);
            acc1 = __builtin_amdgcn_wmma_f32_16x16x4_f32(false, a, false, b1, (short)0, acc1, false, false);
            acc2 = __builtin_amdgcn_wmma_f32_16x16x4_f32(false, a, false, b2, (short)0, acc2, false, false);
            acc3 = __builtin_amdgcn_wmma_f32_16x16x4_f32(false, a, false, b3, (short)0, acc3, false, false);
            a = na; b0 = nb0; b1 = nb1; b2 = nb2; b3 = nb3;
        }

#if USE_TDM
        wait_tensor0();    // this wave's DMA into the next buffer has landed
#elif USE_ASYNC_LDS
        wait_async0();
#endif
        __syncthreads();   // everyone done reading cur / DMA visible to all
    }

    // ---- scatter logits to LDS (reuse buffer 0): VGPR j holds token j+half*8.
    {
        int tokLocal = wave * 16 + half * 8;
        #pragma unroll
        for (int j = 0; j < 8; ++j) {
            float* row = smem + (size_t)(tokLocal + j) * LOGSTRIDE + l16;
            row[0]  = acc0[j];
            row[16] = acc1[j];
            row[32] = acc2[j];
            row[48] = acc3[j];
        }
    }
    __syncthreads();

    // ---- top-2 + softmax, one thread per token (epilogue, negligible cost).
    if (tid < 64) {
        const float* lg = smem + (size_t)tid * LOGSTRIDE;
        float m1 = -1e30f, m2 = -1e30f;
        int   i1 = 0,      i2 = 0;
        #pragma unroll 4
        for (int e = 0; e < NEXP; ++e) {
            float v = lg[e];
            if (v > m1)      { m2 = m1; i2 = i1; m1 = v; i1 = e; }
            else if (v > m2) { m2 = v;  i2 = e; }
        }
        float s = 0.f;
        #pragma unroll 4
        for (int e = 0; e < NEXP; ++e) s += __expf(lg[e] - m1);
        float inv = 1.f / s;
        int g = tok0_wg + tid;
        out_probs[2 * g]     = inv;                   // exp(m1-m1)/s
        out_probs[2 * g + 1] = __expf(m2 - m1) * inv;
        out_idx[2 * g]       = (float)i1;
        out_idx[2 * g + 1]   = (float)i2;
    }
}

extern "C" void kernel_launch(void* const* d_in, const int* in_sizes, int n_in,
                              void* d_out, int out_size, void* d_ws, size_t ws_size,
                              hipStream_t stream) {
    const float* X = (const float*)d_in[0];   // [4,4096,4096] f32
    const float* W = (const float*)d_in[1];   // [64,4096] f32
    int tokens = in_sizes[0] / HIDDEN;        // 16384
    float* probs = (float*)d_out;                          // [tokens,2]
    float* idx   = (float*)d_out + (size_t)tokens * 2;     // [tokens,2]
    int grid = tokens / 64;
    router_topk_kernel<<<grid, 128, SMEM_FLOATS * sizeof(float), stream>>>(
        X, W, probs, idx);
}